// FixedLatentNoiseDecoder_68307159875959
// MI455X (gfx1250) — compile-verified
//
#include <hip/hip_runtime.h>
#include <hip/hip_bf16.h>

typedef __attribute__((ext_vector_type(2))) float v2f;
typedef __attribute__((ext_vector_type(8))) float v8f;

#define TDIM 2048
#define NCH  48          // S*H + 3*N_NOISE = 36 + 12
#define ROWF 9216        // 18*512 floats per (b,t) row of lats
#define NSEG 3072        // 6*512 cols per segment

// ---------------------------------------------------------------------------
// Pass 1: per-batch channel means of x (48 ch) + per-(b,i) gaussian weights.
// One block per batch, 384 threads (48 ch x 8 T-chunks).
// ---------------------------------------------------------------------------
__global__ void flnd_means_kernel(const float* __restrict__ x,
                                  float* __restrict__ means,   // [B,48]
                                  float* __restrict__ wts)     // [B,4,9]
{
    __shared__ float part[384];
    const int b   = blockIdx.x;
    const int tid = threadIdx.x;
    const int c     = tid % 48;
    const int chunk = tid / 48;                 // 0..7
    const float* xb = x + (size_t)b * TDIM * NCH;
    float s = 0.f;
    for (int t = chunk; t < TDIM; t += 8)
        s += xb[(size_t)t * NCH + c];
    part[tid] = s;
    __syncthreads();
    if (tid < 48) {
        float tot = 0.f;
        #pragma unroll
        for (int u = 0; u < 8; ++u) tot += part[u * 48 + tid];
        means[b * 48 + tid] = tot * (1.0f / (float)TDIM);
    }
    if (tid < 4) {
        // smooth channel for noise i = tid is 36 + 3*i + 2
        const int ch = 36 + 3 * tid + 2;
        float tot = 0.f;
        #pragma unroll
        for (int u = 0; u < 8; ++u) tot += part[u * 48 + ch];
        float sig = tot * (1.0f / (float)TDIM);
        sig = fmaxf(sig, 0.001f);
        const float inv2s2 = -0.5f / (sig * sig);
        float w[9], wsum = 0.f;
        #pragma unroll
        for (int j = 0; j < 9; ++j) {
            const float k = (float)(j - 4);
            w[j] = __expf(k * k * inv2s2);
            wsum += w[j];
        }
        const float inv = 1.0f / wsum;
        #pragma unroll
        for (int j = 0; j < 9; ++j)
            wts[(b * 4 + tid) * 9 + j] = w[j] * inv;
    }
}

// ---------------------------------------------------------------------------
// Pass 2: lats GEMM via V_WMMA_F32_16X16X4_F32, fused mean subtraction.
// Grid: B * 3(seg) * 128(t-tiles) * 6(n-blocks of 512). Block = 8 waves;
// each wave computes a 16(T) x 64(N) strip = 4 accumulators x 3 k-steps.
// ---------------------------------------------------------------------------
__global__ void flnd_lat_gemm_kernel(const float* __restrict__ x,
                                     const float* __restrict__ LT,    // [36,18,512]
                                     const float* __restrict__ means, // [B,48]
                                     float* __restrict__ out)
{
    const int lane = threadIdx.x & 31;
    const int wave = threadIdx.x >> 5;

    int blk   = blockIdx.x;
    const int nblk  = blk % 6;   blk /= 6;      // n-block of 512
    const int ttile = blk % 128; blk /= 128;    // t tile of 16
    const int i     = blk % 3;                  // segment
    const int b     = blk / 3;

    const int t0 = ttile * 16;
    const int nb = nblk * 512 + wave * 64;

    const int g  = lane >> 4;      // lane group (0/1)
    const int m  = lane & 15;      // A row / B,C col within tile
    const int k0 = g * 2;          // K element pair owned by this lane group

    // ---- A: env[t0+m, i*12 + h], 3 k-steps of 4 (lane holds K=k0,k0+1) ----
    v2f a[3];
    const float* xrow = x + ((size_t)(b * TDIM + t0 + m)) * NCH + i * 12;
    #pragma unroll
    for (int k = 0; k < 3; ++k) {
        const int h = k * 4 + k0;
        a[k].x = xrow[h];
        a[k].y = xrow[h + 1];
    }

    v8f   c[4] = {v8f{}, v8f{}, v8f{}, v8f{}};
    float meanv[4];
    const float* em = means + b * 48 + i * 12;

    #pragma unroll
    for (int j = 0; j < 4; ++j) {
        const int n = nb + j * 16 + m;
        const float* lcol = LT + (size_t)(i * 12) * ROWF + i * NSEG + n;
        #pragma unroll
        for (int k = 0; k < 3; ++k) {
            const int h = k * 4 + k0;
            v2f bv;
            bv.x = lcol[(size_t)h * ROWF];
            bv.y = lcol[(size_t)(h + 1) * ROWF];
            c[j] = __builtin_amdgcn_wmma_f32_16x16x4_f32(
                       false, a[k], false, bv, (short)0, c[j], false, false);
        }
        // per-column mean of lats over T: einsum(mean_t(env), lat) (linearity)
        float mv = 0.f;
        #pragma unroll
        for (int h = 0; h < 12; ++h)
            mv = fmaf(em[h], lcol[(size_t)h * ROWF], mv);
        meanv[j] = mv;
    }

    // ---- store: C/D layout -> row = t0 + r + 8*g, col = nb + 16*j + m ----
    #pragma unroll
    for (int j = 0; j < 4; ++j) {
        const int n = nb + j * 16 + m;
        #pragma unroll
        for (int r = 0; r < 8; ++r) {
            const int t = t0 + r + 8 * g;
            const size_t idx = (size_t)(b * TDIM + t) * ROWF + i * NSEG + n;
            __builtin_nontemporal_store(c[j][r] - meanv[j], &out[idx]);
        }
    }
}

// ---------------------------------------------------------------------------
// Pass 3: noise i: out[b,t,p] = sum_j w[b,i,j] * (mu[b,t'] + sig[b,t']*eps[b,t',p])
// with t' = reflect(t + j - 4). One thread per output element.
// ---------------------------------------------------------------------------
__global__ void flnd_noise_kernel(const float* __restrict__ x,
                                  const float* __restrict__ eps,
                                  const float* __restrict__ wts,
                                  float* __restrict__ out,
                                  int i, int P, size_t total)
{
    const size_t e = (size_t)blockIdx.x * blockDim.x + threadIdx.x;
    if (e >= total) return;
    const int p = (int)(e % P);
    const int t = (int)((e / P) % TDIM);
    const int b = (int)(e / ((size_t)P * TDIM));

    const float* w  = wts + (b * 4 + i) * 9;
    const int   chm = 36 + 3 * i;

    float acc = 0.f;
    #pragma unroll
    for (int j = 0; j < 9; ++j) {
        int tp = t + j - 4;
        tp = (tp < 0) ? -tp : tp;
        tp = (tp >= TDIM) ? (2 * TDIM - 2 - tp) : tp;
        const float* xr = x + (size_t)(b * TDIM + tp) * NCH + chm;
        const float mu = xr[0];
        const float sg = xr[1];
        const float ep = eps[(size_t)(b * TDIM + tp) * P + p];
        acc = fmaf(w[j], fmaf(sg, ep, mu), acc);
    }
    __builtin_nontemporal_store(acc, &out[e]);
}

// ---------------------------------------------------------------------------
extern "C" void kernel_launch(void* const* d_in, const int* in_sizes, int n_in,
                              void* d_out, int out_size, void* d_ws, size_t ws_size,
                              hipStream_t stream) {
    const float* x  = (const float*)d_in[0];   // [B,2048,48]
    const float* LT = (const float*)d_in[1];   // [36,18,512]
    const float* ep[4] = { (const float*)d_in[2], (const float*)d_in[3],
                           (const float*)d_in[4], (const float*)d_in[5] };
    float* out = (float*)d_out;

    const int B = in_sizes[0] / (TDIM * NCH);

    float* means = (float*)d_ws;               // B*48 floats
    float* wts   = means + (size_t)B * 48;     // B*4*9 floats

    // Pass 1: channel means + gaussian weights
    flnd_means_kernel<<<B, 384, 0, stream>>>(x, means, wts);

    // Pass 2: WMMA GEMM for lats (fully covers B*T*18*512 outputs)
    {
        const int blocks = B * 3 * (TDIM / 16) * 6;   // b * seg * t-tiles * n-blocks
        flnd_lat_gemm_kernel<<<blocks, 256, 0, stream>>>(x, LT, means, out);
    }

    // Pass 3: four noise outputs, concatenated after lats
    size_t off = (size_t)B * TDIM * ROWF;
    for (int i = 0; i < 4; ++i) {
        const int sz = 1 << (i + 2);
        const int P  = sz * sz;
        const size_t total = (size_t)B * TDIM * P;
        const int blocks = (int)((total + 255) / 256);
        flnd_noise_kernel<<<blocks, 256, 0, stream>>>(x, ep[i], wts, out + off, i, P, total);
        off += total;
    }
}